// VectorKuramoto_69965017252365
// MI455X (gfx1250) — compile-verified
//
#include <hip/hip_runtime.h>

// Kuramoto step, trig-factorized into fp32 WMMA GEMMs.
//   coupling[b,i,d] = (1/H) * [ cos(ti)*(M1@sin)[i,d] - sin(ti)*(M1@cos)[i,d]
//                              - cos(ti)*(M2@cos)[i,d] - sin(ti)*(M2@sin)[i,d] ]
//   M1 = relu(0.5*(A+A^T)) .* cos(alpha),  M2 = relu(0.5*(A+A^T)) .* sin(alpha)

#define B_      8
#define H_      1024
#define D_      4
#define TILES_  (H_ / 16)    // 64 row-tiles per batch
#define WAVES_  4
#define BLOCK_  (32 * WAVES_)
#define KSTEPS_ (H_ / 4)     // 256 K-steps of 4 (V_WMMA_F32_16X16X4_F32)

typedef float v2f __attribute__((ext_vector_type(2)));
typedef float v8f __attribute__((ext_vector_type(8)));

__global__ __launch_bounds__(BLOCK_)
void VectorKuramoto_wmma_kernel(const float* __restrict__ theta,
                                const float* __restrict__ gam,
                                const float* __restrict__ A,
                                const float* __restrict__ omega,
                                const float* __restrict__ kappa,
                                const float* __restrict__ alpha,
                                float* __restrict__ out)
{
    __shared__ float ths[H_ * D_];            // sin(theta[b,:,:])  16 KB
    __shared__ float thc[H_ * D_];            // cos(theta[b,:,:])  16 KB
    __shared__ float part1[WAVES_][16 * 16];  // per-wave M1 tile    4 KB
    __shared__ float part2[WAVES_][16 * 16];  // per-wave M2 tile    4 KB

    const int tid  = threadIdx.x;
    const int lane = tid & 31;
    const int wave = tid >> 5;
    const int blk  = blockIdx.x;
    const int b    = blk / TILES_;
    const int i0   = (blk % TILES_) * 16;

    // Stage sin/cos of theta[b,:,:] once per block (global -> LDS).
    // Compile-time stride BLOCK_ => no runtime trip-count division.
    const float* thb = theta + (size_t)b * H_ * D_;
    for (int t = tid; t < H_ * D_; t += BLOCK_) {
        float s, c;
        __sincosf(thb[t], &s, &c);
        ths[t] = s;
        thc[t] = c;
    }
    __syncthreads();

    // wave32 WMMA operand mapping (f32 16x16x4):
    //   A operand: lane (m = lane&15, half = lane>>4), VGPR e -> element (M=m, K=2*half+e)
    //   C/D:       VGPR r, lane -> (M = r + 8*half, N = lane&15)
    const int m     = lane & 15;
    const int khalf = lane >> 4;
    const int row   = i0 + m;

    const float* Ab    = A + (size_t)b * H_ * H_;
    const float* Arow  = Ab + (size_t)row * H_;
    const float* alrow = alpha + (size_t)row * H_;

    v8f acc1 = {};  // M1 @ [sin | cos]
    v8f acc2 = {};  // M2 @ [sin | cos]

#pragma unroll 2
    for (int ks = wave; ks < KSTEPS_; ks += WAVES_) {
        const int k = ks * 4;
        v2f a1, a2, bm;
#pragma unroll
        for (int e = 0; e < 2; ++e) {
            const int kk = khalf * 2 + e;
            const int j  = k + kk;

            // W = relu(0.5*(A + A^T)); row access (consecutive j -> b64-combinable),
            // transpose access coalesced across the 16 lanes of each half.
            float ar = Arow[j];
            float at = Ab[(size_t)j * H_ + row];
            float w  = fmaxf(0.5f * (ar + at), 0.0f);

            float sa, ca;
            __sincosf(alrow[j], &sa, &ca);
            a1[e] = w * ca;   // M1 element (M=row, K=j)
            a2[e] = w * sa;   // M2 element

            // B operand: row K=j, col N=m. cols 0-3 = sin(theta[b,j,d]),
            // cols 4-7 = cos(theta[b,j,d]), cols 8-15 = 0 (selects, no branch: EXEC full)
            float sv = ths[j * D_ + (m & 3)];
            float cv = thc[j * D_ + (m & 3)];
            float v  = (m < 4) ? sv : cv;
            bm[e]    = (m < 8) ? v : 0.0f;
        }
        acc1 = __builtin_amdgcn_wmma_f32_16x16x4_f32(false, a1, false, bm,
                                                     (short)0, acc1, false, false);
        acc2 = __builtin_amdgcn_wmma_f32_16x16x4_f32(false, a2, false, bm,
                                                     (short)0, acc2, false, false);
    }

    // Deterministic cross-wave reduction: staged per-wave tiles, fixed-order sum.
#pragma unroll
    for (int r = 0; r < 8; ++r) {
        const int mm = r + 8 * khalf;
        part1[wave][mm * 16 + m] = acc1[r];
        part2[wave][mm * 16 + m] = acc2[r];
    }
    __syncthreads();

    // Epilogue: 16 rows x D components of the output tile (single pass, 64 lanes).
    for (int t = tid; t < 16 * D_; t += BLOCK_) {
        const int i  = t >> 2;
        const int d  = t & 3;
        const int gi = i0 + i;

        float m1s = 0.f, m1c = 0.f, m2s = 0.f, m2c = 0.f;
#pragma unroll
        for (int w = 0; w < WAVES_; ++w) {
            m1s += part1[w][i * 16 + d];
            m1c += part1[w][i * 16 + d + 4];
            m2s += part2[w][i * 16 + d];
            m2c += part2[w][i * 16 + d + 4];
        }

        const float s  = ths[gi * D_ + d];
        const float c  = thc[gi * D_ + d];
        const float coupling = (1.0f / (float)H_) *
            (c * m1s - s * m1c - c * m2c - s * m2s);

        const float th = thb[gi * D_ + d];
        const float td = omega[gi * D_ + d] + coupling
                       + kappa[gi * D_ + d] * (gam[(size_t)b * H_ + gi] - th);
        out[((size_t)b * H_ + gi) * D_ + d] = th + 1.0f /*DT*/ * td;
    }
}

extern "C" void kernel_launch(void* const* d_in, const int* in_sizes, int n_in,
                              void* d_out, int out_size, void* d_ws, size_t ws_size,
                              hipStream_t stream)
{
    const float* theta = (const float*)d_in[0];  // [B,H,D]
    const float* gam   = (const float*)d_in[1];  // [B,H]
    const float* A     = (const float*)d_in[2];  // [B,H,H]
    const float* omega = (const float*)d_in[3];  // [H,D]
    const float* kappa = (const float*)d_in[4];  // [H,D]
    const float* alpha = (const float*)d_in[5];  // [H,H]
    float* out = (float*)d_out;                  // [B,H,D]

    dim3 grid(B_ * TILES_);   // 512 workgroups, 4 waves each -> 2048 waves
    dim3 block(BLOCK_);
    hipLaunchKernelGGL(VectorKuramoto_wmma_kernel, grid, block, 0, stream,
                       theta, gam, A, omega, kappa, alpha, out);

    (void)in_sizes; (void)n_in; (void)out_size; (void)d_ws; (void)ws_size;
}